// MEAALayer_9414568313194
// MI455X (gfx1250) — compile-verified
//
#include <hip/hip_runtime.h>
#include <math.h>

typedef __attribute__((ext_vector_type(2))) float v2f;
typedef __attribute__((ext_vector_type(8))) float v8f;

#define LN_EPS 1e-5f
#define LEAK   0.2f

#define SLICE 128     // D-columns per histogram workgroup (float)
#define CMAX  128     // max classes for LDS histogram path (64KB static LDS)

// ---------------- zero scratch ----------------
__global__ void k_zero(float* p, size_t n) {
  size_t i = (size_t)blockIdx.x * blockDim.x + threadIdx.x;
  if (i < n) p[i] = 0.f;
}

// ---------------- counts + copy labels to output ----------------
__global__ void k_count(const int* labels, int* counts, int* outLabels, int B) {
  int b = blockIdx.x * blockDim.x + threadIdx.x;
  if (b < B) {
    atomicAdd(&counts[labels[b]], 1);
    outLabels[b] = labels[b];
  }
}

// ---------------- stream features -> output, LDS-accumulated per-class sums ----------------
// block = (m, d-slice, row-chunk). 256 threads = 8 waves; each wave streams one row
// (32 lanes x float4 = SLICE floats), ds_add_f32 into LDS histogram, one flush at end.
__global__ void k_copy_accum_lds(const float* __restrict__ feat, const int* __restrict__ labels,
                                 float* __restrict__ outF, float* __restrict__ sums,
                                 int M, int B, int D, int rows_total, int C,
                                 int S, int CH, int rpc) {
  __shared__ float hist[CMAX * SLICE];
  for (int i = threadIdx.x; i < C * SLICE; i += blockDim.x) hist[i] = 0.f;
  __syncthreads();

  int t = blockIdx.x;
  int chunk = t % CH; t /= CH;
  int slice = t % S;
  int m = t / S;

  int rg = threadIdx.x >> 5;          // wave id in block (8 row streams)
  int ln = threadIdx.x & 31;
  int d0 = slice * SLICE + ln * 4;
  int rowBase = chunk * rpc;
  int rowEnd = rowBase + rpc; if (rowEnd > B) rowEnd = B;

  for (int r = rowBase + rg; r < rowEnd; r += 8) {
    int lab = labels[r];
    const float4 v = *reinterpret_cast<const float4*>(feat + ((size_t)m * B + r) * D + d0);
    __builtin_prefetch(feat + ((size_t)m * B + r + 8) * D + d0, 0, 0);
    *reinterpret_cast<float4*>(outF + ((size_t)m * rows_total + r) * D + d0) = v;
    float* hp = &hist[(lab * SLICE) + ln * 4];
    atomicAdd(hp + 0, v.x); atomicAdd(hp + 1, v.y);   // ds_add_f32
    atomicAdd(hp + 2, v.z); atomicAdd(hp + 3, v.w);
  }
  __syncthreads();

  // flush: one global atomic per (class, column) per workgroup
  for (int i = threadIdx.x; i < C * SLICE; i += blockDim.x) {
    int c = i / SLICE, d = i % SLICE;
    atomicAdd(&sums[((size_t)c * M + m) * D + slice * SLICE + d], hist[i]);
  }
}

// fallback (C > CMAX or D % SLICE != 0): direct global atomics
__global__ void k_copy_accum_direct(const float* __restrict__ feat, const int* __restrict__ labels,
                                    float* __restrict__ outF, float* __restrict__ sums,
                                    int M, int B, int D, int rows_total) {
  int row = blockIdx.x;
  int m = row / B, b = row % B;
  int lab = labels[b];
  int d0 = threadIdx.x * 4;
  const float4 v = *reinterpret_cast<const float4*>(feat + ((size_t)m * B + b) * D + d0);
  *reinterpret_cast<float4*>(outF + ((size_t)m * rows_total + b) * D + d0) = v;
  float* s = sums + ((size_t)lab * M + m) * D + d0;
  atomicAdd(s + 0, v.x); atomicAdd(s + 1, v.y);
  atomicAdd(s + 2, v.z); atomicAdd(s + 3, v.w);
}

// ---------------- proto = sums/cnt (or 0), blend_mask = rowsum > 0 ----------------
__global__ void k_proto(const float* __restrict__ sums, const int* __restrict__ counts,
                        float* __restrict__ proto, float* __restrict__ bmask, int M, int D) {
  int cm = blockIdx.x;
  int c = cm / M, mm = cm % M;
  int cnt = counts[c];
  float inv = (cnt > 0) ? 1.f / (float)((cnt > 1) ? cnt : 1) : 0.f;
  __shared__ float red[256];
  float local = 0.f;
  for (int d = threadIdx.x; d < D; d += blockDim.x) {
    float p = sums[((size_t)c * M + mm) * D + d] * inv;
    proto[((size_t)c * M + mm) * D + d] = p;
    local += p;
  }
  red[threadIdx.x] = local;
  __syncthreads();
  for (int s = 128; s > 0; s >>= 1) {
    if ((int)threadIdx.x < s) red[threadIdx.x] += red[threadIdx.x + s];
    __syncthreads();
  }
  if (threadIdx.x == 0) bmask[c * M + mm] = (red[0] > 0.f) ? 1.f : 0.f;
}

// ---------------- balancing plan: gen/prefix/class_ids + output labels + loss ----------------
__global__ void k_plan(const int* __restrict__ counts, int* __restrict__ class_ids,
                       int* __restrict__ outLabels, float* __restrict__ outF,
                       const float* __restrict__ dw, int B, int C, int desired,
                       size_t lossIdx, int Nhost) {
  __shared__ int offs[1024];
  if (threadIdx.x == 0) {
    int run = 0;
    for (int c = 0; c < C; c++) {
      offs[c] = run;
      int cnt = counts[c];
      int g = (cnt > 0) ? ((desired - cnt > 0) ? desired - cnt : 0) : 0;
      run += g;
    }
  }
  __syncthreads();
  for (int c = threadIdx.x; c < C; c += blockDim.x) {
    int cnt = counts[c];
    int g = (cnt > 0) ? ((desired - cnt > 0) ? desired - cnt : 0) : 0;
    int o = offs[c];
    for (int i = 0; i < g; i++) {
      class_ids[o + i] = c;
      outLabels[B + o + i] = c;
    }
  }
  if (threadIdx.x == 0) outF[lossIdx] = (Nhost > 0) ? 0.1f * dw[0] : 0.0f;
}

// ---------------- base[m,h] = fused0 . W1[m][:D][:,h] ----------------
__global__ void k_base(const float* __restrict__ fused0, const float* __restrict__ W1,
                       float* __restrict__ base, int M, int D, int C, int H) {
  int j = blockIdx.x * blockDim.x + threadIdx.x;
  if (j >= M * H) return;
  int m = j / H, h = j % H;
  const float* w = W1 + (size_t)m * (D + C) * H + h;
  float acc = 0.f;
  for (int d = 0; d < D; d++) acc += fused0[d] * w[(size_t)d * H];
  base[j] = acc;
}

// ---------------- per-class hidden rows: +W1 class row +b1, layernorm, leaky ----------------
__global__ void k_hc(const float* __restrict__ W1, const float* __restrict__ b1,
                     const float* __restrict__ lng, const float* __restrict__ lnb,
                     const float* __restrict__ base, float* __restrict__ hcp,
                     int M, int D, int C, int Cpad, int H) {
  int mc = blockIdx.x;
  int m = mc / Cpad, c = mc % Cpad;
  float* out = hcp + ((size_t)m * Cpad + c) * H;
  if (c >= C) {  // zero padding rows so WMMA tiles are clean
    for (int h = threadIdx.x; h < H; h += blockDim.x) out[h] = 0.f;
    return;
  }
  const float* w = W1 + ((size_t)m * (D + C) + D + c) * H;
  __shared__ float red[256];
  __shared__ float mu_s, rs_s;
  float s = 0.f;
  for (int h = threadIdx.x; h < H; h += blockDim.x) {
    float v = base[(size_t)m * H + h] + w[h] + b1[(size_t)m * H + h];
    out[h] = v;
    s += v;
  }
  red[threadIdx.x] = s; __syncthreads();
  for (int st = 128; st > 0; st >>= 1) {
    if ((int)threadIdx.x < st) red[threadIdx.x] += red[threadIdx.x + st];
    __syncthreads();
  }
  if (threadIdx.x == 0) mu_s = red[0] / (float)H;
  __syncthreads();
  float mu = mu_s;
  float q = 0.f;
  for (int h = threadIdx.x; h < H; h += blockDim.x) {
    float d = out[h] - mu;
    q += d * d;
  }
  __syncthreads();
  red[threadIdx.x] = q; __syncthreads();
  for (int st = 128; st > 0; st >>= 1) {
    if ((int)threadIdx.x < st) red[threadIdx.x] += red[threadIdx.x + st];
    __syncthreads();
  }
  if (threadIdx.x == 0) rs_s = rsqrtf(red[0] / (float)H + LN_EPS);
  __syncthreads();
  float rs = rs_s;
  for (int h = threadIdx.x; h < H; h += blockDim.x) {
    float v = (out[h] - mu) * rs * lng[(size_t)m * H + h] + lnb[(size_t)m * H + h];
    out[h] = (v > 0.f) ? v : LEAK * v;
  }
}

// ---------------- WMMA f32 GEMM, column-panel version ----------------
// one wave owns a full 16-col panel of the output: RT row tiles x 16 cols.
// B (W2) pair loaded once per k-step and reused across RT WMMAs -> W2 read once total.
template <int RT>
__global__ void k_gemm_wmma_panel(const float* __restrict__ hcp, const float* __restrict__ W2,
                                  const float* __restrict__ b2, float* __restrict__ gc,
                                  int M, int Cpad, int H, int D) {
  int wave = (int)((blockIdx.x * blockDim.x + threadIdx.x) >> 5);
  int lane = threadIdx.x & 31;
  int CT = D >> 4;
  int total = M * CT;
  if (wave >= total) return;               // wave-uniform; EXEC stays all-ones
  int m = wave / CT, ct = wave % CT;

  int rl = lane & 15;                      // M index (A) / N index (B,D)
  int kh = (lane >> 4) << 1;               // lanes 16-31 carry K+2,K+3 (ISA 7.12.2)
  const float* Abase = hcp + (size_t)m * Cpad * H + (size_t)rl * H + kh;
  const float* Brow = W2 + (size_t)m * H * D + (size_t)kh * D + ct * 16 + rl;

  v8f acc[RT] = {};
  for (int k = 0; k < H; k += 4) {
    v2f b;
    b.x = Brow[(size_t)k * D];
    b.y = Brow[(size_t)(k + 1) * D];
#pragma unroll
    for (int r = 0; r < RT; r++) {
      const float* Ar = Abase + (size_t)(r * 16) * H + k;
      v2f a;
      a.x = Ar[0];
      a.y = Ar[1];
      acc[r] = __builtin_amdgcn_wmma_f32_16x16x4_f32(false, a, false, b,
                                                     (short)0, acc[r], false, false);
    }
  }
  int col = ct * 16 + rl;
  float bias = b2[(size_t)m * D + col];
  int roff = (lane >= 16) ? 8 : 0;         // C/D layout: lanes 16-31 -> M+8
#pragma unroll
  for (int r = 0; r < RT; r++)
    for (int q = 0; q < 8; q++)
      gc[((size_t)m * Cpad + r * 16 + roff + q) * D + col] = acc[r][q] + bias;
}

// generic fallback: one wave per 16x16 tile
__global__ void k_gemm_wmma_tile(const float* __restrict__ hcp, const float* __restrict__ W2,
                                 const float* __restrict__ b2, float* __restrict__ gc,
                                 int M, int Cpad, int H, int D) {
  int wave = (int)((blockIdx.x * blockDim.x + threadIdx.x) >> 5);
  int lane = threadIdx.x & 31;
  int RT = Cpad >> 4, CT = D >> 4;
  int total = M * RT * CT;
  if (wave >= total) return;
  int m = wave / (RT * CT);
  int rem = wave % (RT * CT);
  int rt = rem / CT, ct = rem % CT;
  int rl = lane & 15;
  int kh = (lane >> 4) << 1;
  const float* Arow = hcp + ((size_t)m * Cpad + rt * 16 + rl) * H + kh;
  const float* Brow = W2 + (size_t)m * H * D + (size_t)kh * D + ct * 16 + rl;
  v8f acc = {};
  for (int k = 0; k < H; k += 4) {
    v2f a, b;
    a.x = Arow[k];
    a.y = Arow[k + 1];
    b.x = Brow[(size_t)k * D];
    b.y = Brow[(size_t)(k + 1) * D];
    acc = __builtin_amdgcn_wmma_f32_16x16x4_f32(false, a, false, b,
                                                (short)0, acc, false, false);
  }
  int col = ct * 16 + rl;
  int rbase = rt * 16 + ((lane >= 16) ? 8 : 0);
  float bias = b2[(size_t)m * D + col];
  for (int r = 0; r < 8; r++)
    gc[((size_t)m * Cpad + rbase + r) * D + col] = acc[r] + bias;
}

// ---------------- per-class sequential prototype blend ----------------
__global__ void k_augc(const float* __restrict__ gc, const float* __restrict__ proto,
                       const float* __restrict__ bmask, const float* __restrict__ corr,
                       float* __restrict__ augc, int M, int C, int Cpad, int D) {
  int mc = blockIdx.x;
  int m = mc / C, c = mc % C;
  for (int d = threadIdx.x; d < D; d += blockDim.x) {
    float v = gc[((size_t)m * Cpad + c) * D + d];
    for (int om = 0; om < M; om++) {
      if (om == m) continue;
      float a = corr[m * M + om] * bmask[c * M + om];
      v = v * (1.f - a) + proto[((size_t)c * M + om) * D + d] * a;
    }
    augc[((size_t)m * C + c) * D + d] = v;
  }
}

// ---------------- scatter per-class rows to the N generated samples ----------------
__global__ void k_scatter(const float* __restrict__ augc, const int* __restrict__ class_ids,
                          float* __restrict__ outF, int M, int B, int N, int C, int D,
                          int rows_total) {
  int t = blockIdx.x;
  int m = t / N, n = t % N;
  int cid = class_ids[n];
  int d0 = threadIdx.x * 4;
  const float4 v = *reinterpret_cast<const float4*>(augc + ((size_t)m * C + cid) * D + d0);
  *reinterpret_cast<float4*>(outF + ((size_t)m * rows_total + B + n) * D + d0) = v;
}

extern "C" void kernel_launch(void* const* d_in, const int* in_sizes, int n_in,
                              void* d_out, int out_size, void* d_ws, size_t ws_size,
                              hipStream_t stream) {
  const float* features = (const float*)d_in[0];
  const float* fused    = (const float*)d_in[1];  // row 0 is all we need
  const int*   labels   = (const int*)  d_in[2];
  const float* W1       = (const float*)d_in[3];
  const float* b1       = (const float*)d_in[4];
  const float* lng      = (const float*)d_in[5];
  const float* lnb      = (const float*)d_in[6];
  const float* W2       = (const float*)d_in[7];
  const float* b2       = (const float*)d_in[8];
  const float* corr     = (const float*)d_in[9];
  const float* dw       = (const float*)d_in[10];

  const int B = in_sizes[2];
  int M = 1;
  while ((long long)(M + 1) * (M + 1) <= (long long)in_sizes[9]) M++;
  const int D = in_sizes[1] / B;
  const int H = in_sizes[4] / M;                                   // 2*D
  const int C = (int)((long long)in_sizes[3] / ((long long)M * H)) - D;
  const int rows_total = (int)(((long long)out_size - 1) / ((long long)M * D + 1));
  const int N = rows_total - B;
  const int desired = (B / C > 1) ? (B / C) : 1;
  const int Cpad = ((C + 15) / 16) * 16;

  // ---- workspace carve-up ----
  char* ws = (char*)d_ws;
  size_t off = 0;
  auto carve = [&](size_t bytes) { size_t r = off; off = (off + bytes + 255) & ~(size_t)255; return r; };
  int*   counts    = (int*)  (ws + carve((size_t)C * 4));
  int*   class_ids = (int*)  (ws + carve((size_t)B * 4));
  float* sums      = (float*)(ws + carve((size_t)C * M * D * 4));
  float* proto     = (float*)(ws + carve((size_t)C * M * D * 4));
  float* bmask     = (float*)(ws + carve((size_t)C * M * 4));
  float* base      = (float*)(ws + carve((size_t)M * H * 4));
  float* hcp       = (float*)(ws + carve((size_t)M * Cpad * H * 4));
  float* gc        = (float*)(ws + carve((size_t)M * Cpad * D * 4));
  float* augc      = (float*)(ws + carve((size_t)M * C * D * 4));

  float* outF = (float*)d_out;
  int* outLabels = (int*)(outF + (size_t)M * rows_total * D);
  size_t lossIdx = (size_t)M * rows_total * D + rows_total;  // float index (int area aliases)

  // zero counts + sums
  {
    size_t n = (size_t)C;
    k_zero<<<(unsigned)((n + 255) / 256), 256, 0, stream>>>((float*)counts, n);
    n = (size_t)C * M * D;
    k_zero<<<(unsigned)((n + 255) / 256), 256, 0, stream>>>(sums, n);
  }

  // counts + label passthrough
  k_count<<<(B + 255) / 256, 256, 0, stream>>>(labels, counts, outLabels, B);

  // stream features -> output; per-class sums via LDS histogram (ds_add_f32) when it fits
  if (C <= CMAX && (D % SLICE) == 0) {
    const int S = D / SLICE;
    const int CH = 16;                              // row chunks
    const int rpc = (B + CH - 1) / CH;
    k_copy_accum_lds<<<M * S * CH, 256, 0, stream>>>(features, labels, outF, sums,
                                                     M, B, D, rows_total, C, S, CH, rpc);
  } else {
    k_copy_accum_direct<<<M * B, D / 4, 0, stream>>>(features, labels, outF, sums,
                                                     M, B, D, rows_total);
  }

  // proto + blend mask
  k_proto<<<C * M, 256, 0, stream>>>(sums, counts, proto, bmask, M, D);

  // balancing plan + generated labels + loss scalar
  k_plan<<<1, 256, 0, stream>>>(counts, class_ids, outLabels, outF, dw, B, C, desired,
                                lossIdx, N);

  if (N > 0) {
    // shared GEMV: fused0 . W1[:, :D, :]
    k_base<<<(M * H + 255) / 256, 256, 0, stream>>>(fused, W1, base, M, D, C, H);

    // per-class hidden rows (+ LN + leaky), padded to Cpad
    k_hc<<<M * Cpad, 256, 0, stream>>>(W1, b1, lng, lnb, base, hcp, M, D, C, Cpad, H);

    // WMMA f32 GEMM: gc = hcp @ W2 + b2 (column-panel waves, W2 read once)
    {
      int RT = Cpad / 16;
      int CT = D / 16;
      int waves = M * CT;
      int blocks = (waves + 3) / 4;                 // 4 waves / 128-thread block
      switch (RT) {
        case 1: k_gemm_wmma_panel<1><<<blocks, 128, 0, stream>>>(hcp, W2, b2, gc, M, Cpad, H, D); break;
        case 2: k_gemm_wmma_panel<2><<<blocks, 128, 0, stream>>>(hcp, W2, b2, gc, M, Cpad, H, D); break;
        case 3: k_gemm_wmma_panel<3><<<blocks, 128, 0, stream>>>(hcp, W2, b2, gc, M, Cpad, H, D); break;
        case 4: k_gemm_wmma_panel<4><<<blocks, 128, 0, stream>>>(hcp, W2, b2, gc, M, Cpad, H, D); break;
        case 5: k_gemm_wmma_panel<5><<<blocks, 128, 0, stream>>>(hcp, W2, b2, gc, M, Cpad, H, D); break;
        case 6: k_gemm_wmma_panel<6><<<blocks, 128, 0, stream>>>(hcp, W2, b2, gc, M, Cpad, H, D); break;
        case 7: k_gemm_wmma_panel<7><<<blocks, 128, 0, stream>>>(hcp, W2, b2, gc, M, Cpad, H, D); break;
        case 8: k_gemm_wmma_panel<8><<<blocks, 128, 0, stream>>>(hcp, W2, b2, gc, M, Cpad, H, D); break;
        default: {
          int tiles = M * RT * CT;
          k_gemm_wmma_tile<<<(tiles + 3) / 4, 128, 0, stream>>>(hcp, W2, b2, gc, M, Cpad, H, D);
        } break;
      }
    }

    // sequential prototype blending per class
    k_augc<<<M * C, 256, 0, stream>>>(gc, proto, bmask, corr, augc, M, C, Cpad, D);

    // scatter class rows to the N generated samples
    k_scatter<<<M * N, D / 4, 0, stream>>>(augc, class_ids, outF, M, B, N, C, D, rows_total);
  }
}